// PostProcessor_46866683134501
// MI455X (gfx1250) — compile-verified
//
#include <hip/hip_runtime.h>
#include <math.h>

// ---------------------------------------------------------------------------
// Rotated-box detection post-processor for MI455X (gfx1250, wave32).
//
// Heavy cost = rotated IoU (Sutherland-Hodgman quad clip). We eliminate ~90%
// of clips with a circumcircle prefilter computed on the matrix cores:
//   dist^2 - (ri+rj)^2  =  [xi yi ri si] x [-2xj -2yj -2rj 1]^T + sj
// which maps 1:1 onto V_WMMA_F32_16X16X4_F32 (16x4 * 4x16 -> 16x16 f32).
// The 16x16 D tile is bit-packed per lane (each lane owns 8 row-bits of one
// column) and stored transposed with one ds_store_b8 per lane -- no ballots,
// no predicated stores.
// ---------------------------------------------------------------------------

#define NPROP      512
#define NCLS       16
#define NFG        15
#define REG        5
#define SCORE_TH   0.05f
#define NMS_TH     0.5f
#define DETS       100
#define CSTRIDE    128                  // compact slots per (image,class)
#define XFORM_CLIP 4.135166556742356f   // log(1000/16)

typedef __attribute__((ext_vector_type(2))) float f32x2;
typedef __attribute__((ext_vector_type(8))) float f32x8;

// ---- geometry helpers ------------------------------------------------------

__device__ __forceinline__ void rot_corners(float xc, float yc, float w, float h,
                                            float cs, float sn,
                                            float* cx, float* cy) {
  const float w2 = 0.5f * w, h2 = 0.5f * h;
  const float lx[4] = {-w2, w2, w2, -w2};
  const float ly[4] = {-h2, -h2, h2, h2};
#pragma unroll
  for (int k = 0; k < 4; ++k) {
    cx[k] = xc + lx[k] * cs - ly[k] * sn;
    cy[k] = yc + lx[k] * sn + ly[k] * cs;
  }
}

// Sutherland-Hodgman: clip quad A by CCW quad B, return intersection area.
__device__ float quad_inter_area(const float* ax, const float* ay,
                                 const float* bx4, const float* by4) {
  float px[8] = {0}, py[8] = {0}, qx[8] = {0}, qy[8] = {0};
  int n = 4;
#pragma unroll
  for (int k = 0; k < 4; ++k) { px[k] = ax[k]; py[k] = ay[k]; }
  for (int e = 0; e < 4; ++e) {
    const float p1x = bx4[e], p1y = by4[e];
    const float p2x = bx4[(e + 1) & 3], p2y = by4[(e + 1) & 3];
    const float ex = p2x - p1x, ey = p2y - p1y;
    int m = 0;
    for (int k = 0; k < n; ++k) {
      const int kn = (k + 1 < n) ? k + 1 : 0;
      const float cx = px[k], cy = py[k], nx = px[kn], ny = py[kn];
      const float sc  = ex * (cy - p1y) - ey * (cx - p1x);
      const float snn = ex * (ny - p1y) - ey * (nx - p1x);
      const bool inc = (sc >= 0.0f), inn = (snn >= 0.0f);
      if (inc) { if (m < 8) { qx[m] = cx; qy[m] = cy; } ++m; }
      if (inc != inn) {
        const float den = sc - snn;
        const float t = sc / (fabsf(den) > 1e-8f ? den : 1.0f);
        if (m < 8) { qx[m] = cx + t * (nx - cx); qy[m] = cy + t * (ny - cy); }
        ++m;
      }
    }
    n = (m < 8) ? m : 8;
#pragma unroll
    for (int k = 0; k < 8; ++k) { px[k] = qx[k]; py[k] = qy[k]; }
  }
  if (n < 3) return 0.0f;
  float area = 0.0f;
  for (int k = 0; k < n; ++k) {
    const int kn = (k + 1 < n) ? k + 1 : 0;
    area += px[k] * py[kn] - px[kn] * py[k];
  }
  return 0.5f * fabsf(area);
}

// ---- per-(image,class) kernel: softmax+decode, sort, WMMA prefilter, NMS ---

__global__ __launch_bounds__(NPROP)
void rnms_class_kernel(const float* __restrict__ logits,
                       const float* __restrict__ boxreg,
                       const float* __restrict__ props,
                       float* __restrict__ cScore,   // [B][NFG][CSTRIDE]
                       float* __restrict__ cBox) {   // [B][NFG][CSTRIDE][5]
  const int b   = blockIdx.x / NFG;
  const int c   = blockIdx.x % NFG;     // fg class index, real class = c+1
  const int cls = c + 1;
  const int tid  = threadIdx.x;
  const int lane = tid & 31;
  const int wave = tid >> 5;

  __shared__ float sscore[NPROP];
  __shared__ int   sidx[NPROP];
  __shared__ float bxs[NPROP], bys[NPROP], bws[NPROP], bhs[NPROP], bas[NPROP];
  __shared__ float bcs[NPROP], bsn[NPROP], barea[NPROP], brad[NPROP], bsum[NPROP];
  // transposed prefilter mask: smaskT[j][i>>3] bit (i&7)  (32 KB)
  __shared__ unsigned char smaskT[NPROP * (NPROP / 8)];
  __shared__ unsigned int  skeep[NPROP / 32];

  // 1) softmax score for this class
  {
    const float* lg = logits + ((size_t)(b * NPROP + tid)) * NCLS;
    float mx = lg[0];
#pragma unroll
    for (int k = 1; k < NCLS; ++k) mx = fmaxf(mx, lg[k]);
    float sum = 0.0f;
#pragma unroll
    for (int k = 0; k < NCLS; ++k) sum += expf(lg[k] - mx);
    const float p = expf(lg[cls] - mx) / sum;
    sscore[tid] = (p > SCORE_TH) ? p : -INFINITY;
    sidx[tid]   = tid;
  }
  __syncthreads();

  // 2) bitonic sort: score descending, original index ascending on ties
  for (unsigned k = 2; k <= NPROP; k <<= 1) {
    for (unsigned j = k >> 1; j > 0; j >>= 1) {
      const unsigned i = tid, ixj = i ^ j;
      if (ixj > i) {
        const float s1 = sscore[i], s2 = sscore[ixj];
        const int   i1 = sidx[i],   i2 = sidx[ixj];
        const bool b21 = (s2 > s1) || (s2 == s1 && i2 < i1); // e[ixj] before e[i]
        const bool b12 = (s1 > s2) || (s1 == s2 && i1 < i2);
        if (((i & k) == 0) ? b21 : b12) {
          sscore[i] = s2; sscore[ixj] = s1;
          sidx[i]   = i2; sidx[ixj]   = i1;
        }
      }
      __syncthreads();
    }
  }

  // 3) decode box for my sorted slot; build box info + prefilter features
  {
    const int o = sidx[tid];
    const float* rg = boxreg + ((size_t)(b * NPROP + o)) * (NCLS * REG) + cls * REG;
    const float* pr = props  + ((size_t)(b * NPROP + o)) * REG;
    const float dx = rg[0] * 0.1f, dy = rg[1] * 0.1f;
    const float dw = fminf(rg[2] * 0.2f, XFORM_CLIP);
    const float dh = fminf(rg[3] * 0.2f, XFORM_CLIP);
    const float da = rg[4] * 0.1f;
    const float xc = pr[0], yc = pr[1], w = pr[2], h = pr[3], a = pr[4];
    const float px = dx * w + xc, py = dy * h + yc;
    const float pw = expf(dw) * w, ph = expf(dh) * h;
    const float pa = da * 57.29577951308232f + a;
    bxs[tid] = px; bys[tid] = py; bws[tid] = pw; bhs[tid] = ph; bas[tid] = pa;
    const float t = pa * 0.017453292519943295f;
    bcs[tid] = cosf(t); bsn[tid] = sinf(t);
    barea[tid] = pw * ph;
    const float r = 0.5f * sqrtf(pw * pw + ph * ph);
    brad[tid] = r;
    bsum[tid] = px * px + py * py - r * r;
    // keep-mask init from validity (sorted order)
    const unsigned bal = __builtin_amdgcn_ballot_w32(sscore[tid] > SCORE_TH);
    if (lane == 0) skeep[wave] = bal;
  }
  __syncthreads();

  // 4) WMMA circumcircle prefilter: 32x32 tiles of 16x16, 1024 tiles / 16 waves.
  //    D[i][j] + sj = dist^2 - (ri+rj)^2 ; pass iff <= margin.
  //    Lane l of the D tile holds rows i0+v(+8 for l>=16) of column j0+(l&15),
  //    so each lane packs its 8 row-bits into one byte of the transposed mask.
  for (int t = wave; t < (NPROP / 16) * (NPROP / 16); t += NPROP / 32) {
    const int ti = t >> 5, tj = t & 31;
    const int i0 = ti * 16, j0 = tj * 16;
    const int lh = lane & 15;
    f32x2 A, B2;
    if (lane < 16) {
      A[0]  = bxs[i0 + lh];          A[1]  = bys[i0 + lh];
      B2[0] = -2.0f * bxs[j0 + lh];  B2[1] = -2.0f * bys[j0 + lh];
    } else {
      A[0]  = brad[i0 + lh];         A[1]  = bsum[i0 + lh];
      B2[0] = -2.0f * brad[j0 + lh]; B2[1] = 1.0f;
    }
    f32x8 acc = {};
    acc = __builtin_amdgcn_wmma_f32_16x16x4_f32(
        false, A, false, B2, (short)0, acc, false, false);
    const float sj = bsum[j0 + lh];
    unsigned bits = 0u;
#pragma unroll
    for (int v = 0; v < 8; ++v)
      bits |= ((acc[v] + sj <= 1.0f) ? 1u : 0u) << v;   // 1 px^2 margin
    // lanes 0-15: rows i0..i0+7 -> byte 2*ti ; lanes 16-31: rows i0+8..+15
    smaskT[(j0 + lh) * (NPROP / 8) + 2 * ti + (lane >> 4)] = (unsigned char)bits;
  }
  __syncthreads();

  // 5) greedy NMS: thread tid owns candidate tid; exact clip only where the
  //    WMMA prefilter says circumcircles overlap.
  float jx[4], jy[4];
  rot_corners(bxs[tid], bys[tid], bws[tid], bhs[tid], bcs[tid], bsn[tid], jx, jy);
  const float areaJ = barea[tid];
  __syncthreads();

  for (int i = 0; i < NPROP - 1; ++i) {
    const bool alive = (skeep[i >> 5] >> (i & 31)) & 1u;
    if (alive && tid > i) {
      const bool kj = (skeep[tid >> 5] >> (tid & 31)) & 1u;
      const unsigned char mb = smaskT[tid * (NPROP / 8) + (i >> 3)];
      if (kj && ((mb >> (i & 7)) & 1u)) {
        float ix4[4], iy4[4];
        rot_corners(bxs[i], bys[i], bws[i], bhs[i], bcs[i], bsn[i], ix4, iy4);
        const float inter = quad_inter_area(ix4, iy4, jx, jy);
        const float uni = fmaxf(barea[i] + areaJ - inter, 1e-8f);
        if (inter > NMS_TH * uni)
          atomicAnd(&skeep[tid >> 5], ~(1u << (tid & 31)));
      }
    }
    __syncthreads();
  }

  // 6) rank kept boxes (prefix popcount), cap at DETS, compact to workspace
  const bool kp = (skeep[tid >> 5] >> (tid & 31)) & 1u;
  int rank = 0;
  for (int w = 0; w < NPROP / 32; ++w) {
    const unsigned word = skeep[w];
    if (w < (tid >> 5))       rank += __popc(word);
    else if (w == (tid >> 5)) rank += __popc(word & ((1u << (tid & 31)) - 1u));
  }
  float* csc = cScore + (size_t)(b * NFG + c) * CSTRIDE;
  float* cbx = cBox   + (size_t)(b * NFG + c) * CSTRIDE * REG;
  if (tid < CSTRIDE) csc[tid] = -INFINITY;
  __syncthreads();
  if (kp && rank < DETS) {
    csc[rank] = sscore[tid];
    float* o = cbx + rank * REG;
    o[0] = bxs[tid]; o[1] = bys[tid]; o[2] = bws[tid]; o[3] = bhs[tid]; o[4] = bas[tid];
  }
}

// ---- per-image cross-class top-100 -----------------------------------------

__global__ __launch_bounds__(512)
void rnms_topk_kernel(const float* __restrict__ cScore,
                      const float* __restrict__ cBox,
                      float* __restrict__ out) {
  const int b = blockIdx.x;
  const int tid = threadIdx.x;
  __shared__ unsigned long long keys[2048];

  // pack monotone keys: hi32 = order-preserving float bits, lo32 = ~(c<<16|r)
  for (int e = tid; e < 2048; e += blockDim.x) {
    unsigned long long key = 0ull;
    if (e < NFG * DETS) {
      const int c = e / DETS, r = e % DETS;
      const float s = cScore[(size_t)(b * NFG + c) * CSTRIDE + r];
      const unsigned u = __float_as_uint(s);
      const unsigned m = (u & 0x80000000u) ? ~u : (u | 0x80000000u);
      const unsigned lo = ~(((unsigned)c << 16) | (unsigned)r);
      key = ((unsigned long long)m << 32) | lo;
    }
    keys[e] = key;
  }
  __syncthreads();

  // bitonic sort descending (ties -> ascending (class, rank) == flat index)
  for (unsigned k = 2; k <= 2048; k <<= 1) {
    for (unsigned j = k >> 1; j > 0; j >>= 1) {
      for (unsigned i = tid; i < 2048; i += blockDim.x) {
        const unsigned ixj = i ^ j;
        if (ixj > i) {
          const unsigned long long a = keys[i], bb = keys[ixj];
          const bool sw = ((i & k) == 0) ? (a < bb) : (a > bb);
          if (sw) { keys[i] = bb; keys[ixj] = a; }
        }
      }
      __syncthreads();
    }
  }

  if (tid < DETS) {
    const unsigned long long key = keys[tid];
    const unsigned m = (unsigned)(key >> 32);
    const unsigned u = (m & 0x80000000u) ? (m ^ 0x80000000u) : ~m;
    const float s = __uint_as_float(u);
    float* o = out + (size_t)(b * DETS + tid) * 6;
    const bool ok = isfinite(s);
    if (ok) {
      const unsigned cr = ~((unsigned)(key & 0xffffffffu));
      const int c = (int)(cr >> 16), r = (int)(cr & 0xffffu);
      const float* bp = cBox + ((size_t)(b * NFG + c) * CSTRIDE + r) * REG;
      o[0] = bp[0]; o[1] = bp[1]; o[2] = bp[2]; o[3] = bp[3]; o[4] = bp[4];
      o[5] = s;
    } else {
      o[0] = 0.f; o[1] = 0.f; o[2] = 0.f; o[3] = 0.f; o[4] = 0.f; o[5] = 0.f;
    }
  }
}

// ---- launch ----------------------------------------------------------------

extern "C" void kernel_launch(void* const* d_in, const int* in_sizes, int n_in,
                              void* d_out, int out_size, void* d_ws, size_t ws_size,
                              hipStream_t stream) {
  (void)n_in; (void)out_size; (void)ws_size;
  const float* logits = (const float*)d_in[0];   // [B,512,16]
  const float* boxreg = (const float*)d_in[1];   // [B,512,80]
  const float* props  = (const float*)d_in[2];   // [B,512,5]
  float* out = (float*)d_out;                    // [B,100,6]
  const int B = in_sizes[0] / (NPROP * NCLS);

  float* cScore = (float*)d_ws;                       // B*NFG*CSTRIDE floats
  float* cBox   = cScore + (size_t)B * NFG * CSTRIDE; // B*NFG*CSTRIDE*5 floats

  rnms_class_kernel<<<dim3(B * NFG), dim3(NPROP), 0, stream>>>(
      logits, boxreg, props, cScore, cBox);
  rnms_topk_kernel<<<dim3(B), dim3(512), 0, stream>>>(cScore, cBox, out);
}